// EmbeddingToProbability_26989574488092
// MI455X (gfx1250) — compile-verified
//
#include <hip/hip_runtime.h>

typedef __attribute__((ext_vector_type(2))) float v2f;
typedef __attribute__((ext_vector_type(8))) float v8f;

#define XYZ (128 * 128 * 64)   // 1,048,576 voxels
#define NC  96                 // centroids

// out[n, v] = exp(-sum_k (e_k[v] - c_k[n])^2 / (2 sigma_k^2))
// Rewritten as D = A x B + C with V_WMMA_F32_16X16X4_F32:
//   A[m][:] = (e0, e1, e2, q_m)        q_m = -sum_k e_k^2 * p_k
//   B[:][n] = (g0n, g1n, g2n, 1)       g_kn = 2 * p_k * c_kn
//   C[m][n] = r_n                      r_n  = -sum_k c_kn^2 * p_k
//   p_k = log2(e) / (2 sigma_k^2)      out = exp2(D)
__global__ __launch_bounds__(256) void e2p_wmma_kernel(
    const float* __restrict__ emb,    // [3, XYZ]
    const float* __restrict__ cent,   // [NC, 3]
    const float* __restrict__ sigma,  // [3]
    const float* __restrict__ scale,  // [3]
    float* __restrict__ out)          // [NC, XYZ]
{
    __shared__ float sB[NC * 4];   // (g0, g1, g2, 1) per centroid
    __shared__ float sR[NC];       // r_n per centroid
    __shared__ float sred[256];

    const int t    = threadIdx.x;
    const int lane = t & 31;
    const int m    = lane & 15;     // row-in-tile (voxel for A, centroid for B/D)
    const int half = lane >> 4;     // 0: lanes 0-15, 1: lanes 16-31
    const int wave = t >> 5;        // 0..7

    // ---- per-block centroid-constant precompute (cheap: 96x5 floats) ----
    float mx = -3.0e38f;
    for (int i = t; i < NC * 3; i += 256) mx = fmaxf(mx, cent[i]);
    sred[t] = mx;
    __syncthreads();
    for (int s = 128; s > 0; s >>= 1) {
        if (t < s) sred[t] = fmaxf(sred[t], sred[t + s]);
        __syncthreads();
    }
    const float cmax = sred[0];

    const float LOG2E = 1.44269504088896340736f;
    const float s0 = sigma[0] + 1e-10f;
    const float s1 = sigma[1] + 1e-10f;
    const float s2 = sigma[2] + 1e-10f;
    const float p0 = LOG2E / (2.0f * s0 * s0);
    const float p1 = LOG2E / (2.0f * s1 * s1);
    const float p2 = LOG2E / (2.0f * s2 * s2);

    if (t < NC) {
        float c0 = cent[t * 3 + 0];
        float c1 = cent[t * 3 + 1];
        float c2 = cent[t * 3 + 2];
        if (cmax > 5.0f) { c0 /= scale[0]; c1 /= scale[1]; c2 /= scale[2]; }
        sB[t * 4 + 0] = 2.0f * p0 * c0;
        sB[t * 4 + 1] = 2.0f * p1 * c1;
        sB[t * 4 + 2] = 2.0f * p2 * c2;
        sB[t * 4 + 3] = 1.0f;
        sR[t] = -(c0 * c0 * p0 + c1 * c1 * p1 + c2 * c2 * p2);
    }
    __syncthreads();

    // ---- A operand: 16 contiguous voxels per wave ----
    const int vbase = (blockIdx.x * 8 + wave) * 16;
    const int v     = vbase + m;
    const float e0 = emb[v];
    const float e1 = emb[XYZ + v];
    const float e2 = emb[2 * XYZ + v];
    const float q  = -(e0 * e0 * p0 + e1 * e1 * p1 + e2 * e2 * p2);

    // 16x4 f32 A layout: lanes 0-15 -> (K0,K1), lanes 16-31 -> (K2,K3)
    v2f a;
    a.x = half ? e2 : e0;
    a.y = half ? q  : e1;

    const float* bptr = &sB[m * 4 + half * 2];   // 8-byte aligned, conflict-free

    #pragma unroll
    for (int nt = 0; nt < 6; ++nt) {             // 6 centroid tiles of 16
        // 4x16 f32 B layout: V0 = rows K0(lo)/K2(hi), V1 = rows K1(lo)/K3(hi)
        v2f b;
        b.x = bptr[nt * 64 + 0];
        b.y = bptr[nt * 64 + 1];

        const float r = sR[nt * 16 + m];         // C[m][n] = r_n, n = lane&15
        v8f c = { r, r, r, r, r, r, r, r };

        v8f d = __builtin_amdgcn_wmma_f32_16x16x4_f32(
            /*neg_a=*/false, a, /*neg_b=*/false, b,
            /*c_mod=*/(short)0, c, /*reuse_a=*/false, /*reuse_b=*/false);

        // D: vgpr i, lane -> voxel (vbase + i + 8*half), centroid (nt*16 + m)
        float* o = out + (size_t)(nt * 16 + m) * XYZ + vbase + half * 8;
        #pragma unroll
        for (int i = 0; i < 8; ++i) {
            o[i] = __builtin_amdgcn_exp2f(d[i]);   // v_exp_f32 (base-2 folded in p_k)
        }
    }
}

extern "C" void kernel_launch(void* const* d_in, const int* in_sizes, int n_in,
                              void* d_out, int out_size, void* d_ws, size_t ws_size,
                              hipStream_t stream) {
    (void)in_sizes; (void)n_in; (void)d_ws; (void)ws_size; (void)out_size;
    const float* emb   = (const float*)d_in[0];
    const float* cent  = (const float*)d_in[1];
    const float* sigma = (const float*)d_in[2];
    const float* scale = (const float*)d_in[3];
    float* out = (float*)d_out;

    // 8192 blocks x 256 threads: 8 waves/block, 16 voxels/wave -> exact cover
    e2p_wmma_kernel<<<dim3(XYZ / 128), dim3(256), 0, stream>>>(
        emb, cent, sigma, scale, out);
}